// MeanAggregator_40355512713735
// MI455X (gfx1250) — compile-verified
//
#include <hip/hip_runtime.h>

typedef __attribute__((ext_vector_type(2))) float        v2f;
typedef __attribute__((ext_vector_type(8))) float        v8f;
typedef __attribute__((ext_vector_type(4))) unsigned int v4u;
typedef __attribute__((ext_vector_type(8))) int          v8i;
typedef __attribute__((ext_vector_type(4))) int          v4i;

#define D_FEAT    128
#define OP_ROWS   8            // rows per TDM gather op (32-bit index mode max)
#define TILE_E    64           // edges per LDS tile = 8 waves x 8 rows
#define ROW_PITCH 516          // 128 dwords + 1 pad dword (TDM pad_enable)

__device__ __forceinline__ int lower_bound_i32(const int* __restrict__ a, int n, int key) {
    int lo = 0, hi = n;
    while (lo < hi) {
        int mid = (lo + hi) >> 1;
        if (a[mid] < key) lo = mid + 1; else hi = mid;
    }
    return lo;
}

// One workgroup (8 wave32) per 16-row output tile. Edge span for the tile is
// contiguous (row_ids sorted). Feature rows are gathered 8-at-a-time into LDS
// by the Tensor Data Mover (gather mode, 32-bit indices), double-buffered on
// TENSORcnt. Segment-sum is D(16x16) += A(16x4 membership) x B(4x16 features)
// via V_WMMA_F32_16X16X4_F32; wave w owns feature columns [16w,16w+16).
__global__ __launch_bounds__(256, 4)
void mean_agg_tdm_wmma(const float* __restrict__ features,
                       const int*   __restrict__ neigh_ids,
                       const int*   __restrict__ row_ids,
                       float*       __restrict__ out,
                       int n_edges, int n_rows, int n_nodes) {
    __shared__ int s_off[17];
    __shared__ int s_rows[2][TILE_E];
    __shared__ __align__(16) char s_feat[2][TILE_E * ROW_PITCH];

    const int base = blockIdx.x * 16;
    const int t    = threadIdx.x;

    if (t < 17) {
        int key = base + t;
        if (key > n_rows) key = n_rows;
        s_off[t] = lower_bound_i32(row_ids, n_edges, key);
    }
    __syncthreads();

    const int e0      = s_off[0];
    const int e1      = s_off[16];
    const int wave    = t >> 5;
    const int lane    = t & 31;
    const int half    = lane >> 4;
    const int nn      = lane & 15;      // M for A, N for B/D
    const int colbase = wave * 16;
    const int kA      = 2 * half;       // K of vector slot 0

    v8f acc = {0.f, 0.f, 0.f, 0.f, 0.f, 0.f, 0.f, 0.f};

    const int nspan = e1 - e0;
    if (nspan > 0) {
        const unsigned e1u  = (unsigned)e1;
        const unsigned e1m1 = (unsigned)(e1 - 1);
        const int ntiles    = (nspan + TILE_E - 1) / TILE_E;

        const unsigned long long fbase = (unsigned long long)(uintptr_t)features;
        const unsigned lds0 = (unsigned)(uintptr_t)(void*)&s_feat[0][0];
        const unsigned lds1 = (unsigned)(uintptr_t)(void*)&s_feat[1][0];

        // Issue one TDM gather (8 feature rows -> LDS) per wave for one tile,
        // and stage the tile's row_ids into LDS.
        auto issue_tile = [&](int tileIdx, int bufIdx) {
            const int ecb = e0 + tileIdx * TILE_E;
            if (t < TILE_E) {
                unsigned ev = (unsigned)(ecb + t);
                s_rows[bufIdx][t] = row_ids[ev < e1u ? ev : e1m1];
            }
            const int eb = ecb + wave * OP_ROWS;
            __builtin_prefetch(neigh_ids + eb + 2 * TILE_E, 0, 1);
            int idx[OP_ROWS];
#pragma unroll
            for (int i = 0; i < OP_ROWS; ++i) {
                unsigned ev = (unsigned)(eb + i);
                idx[i] = neigh_ids[ev < e1u ? ev : e1m1];
            }
            // ---- D# group 0: flags | lds_addr | global_addr | type ----
            v4u g0;
            g0.x = 0xC0000001u;  // count=1, gather_mode=1, 32-bit indices
            g0.y = (bufIdx ? lds1 : lds0) + (unsigned)(wave * OP_ROWS * ROW_PITCH);
            g0.z = (unsigned)(fbase & 0xFFFFFFFFull);
            g0.w = (unsigned)((fbase >> 32) & 0x1FFFFFFull) | (2u << 30); // type=2
            // ---- D# group 1 ----
            v8i g1;
            g1[0] = (2 << 16) | (1 << 20) | (6 << 22); // data_size=4B, pad 1dw/128dw
            g1[1] = (int)(128u << 16);                 // tensor_dim0 = 128 (lo16)
            g1[2] = (int)(((unsigned)n_nodes & 0xFFFFu) << 16); // dim0 hi=0 | dim1 lo16
            g1[3] = (int)((128u << 16) | ((unsigned)n_nodes >> 16)); // tile_dim0|dim1 hi
            g1[4] = OP_ROWS;                           // tile_dim1 = #indices
            g1[5] = D_FEAT;                            // tensor_dim0_stride (elements)
            g1[6] = 0;
            g1[7] = 0;
            // ---- D# groups 2/3: gather row indices (32-bit mode) ----
            v4i g2, g3;
            g2[0] = idx[0]; g2[1] = idx[1]; g2[2] = idx[2]; g2[3] = idx[3];
            g3[0] = idx[4]; g3[1] = idx[5]; g3[2] = idx[6]; g3[3] = idx[7];
            // 6-arg toolchain form: extra int32x8 operand (zero-filled), cpol=0
            v8i gz = {0, 0, 0, 0, 0, 0, 0, 0};
            __builtin_amdgcn_tensor_load_to_lds(g0, g1, g2, g3, gz, 0);
        };

        issue_tile(0, 0);
        for (int ti = 0; ti < ntiles; ++ti) {
            const int bi = ti & 1;
            if (ti + 1 < ntiles) {
                issue_tile(ti + 1, (ti + 1) & 1);
                __builtin_amdgcn_s_wait_tensorcnt(1);  // tile ti landed
            } else {
                __builtin_amdgcn_s_wait_tensorcnt(0);
            }
            __syncthreads();   // all 8 waves' TDM slices visible

            const int  ecb = e0 + ti * TILE_E;
            const char* fb = s_feat[bi];
            const int* rw  = s_rows[bi];
#pragma unroll
            for (int c = 0; c < TILE_E / 4; ++c) {
                const int le0 = 4 * c + kA;
                const int le1 = le0 + 1;
                const float b0 = *(const float*)(fb + le0 * ROW_PITCH + (colbase + nn) * 4);
                const float b1 = *(const float*)(fb + le1 * ROW_PITCH + (colbase + nn) * 4);
                const float a0 = ((ecb + le0 < e1) && (rw[le0] == base + nn)) ? 1.0f : 0.0f;
                const float a1 = ((ecb + le1 < e1) && (rw[le1] == base + nn)) ? 1.0f : 0.0f;
                v2f A = {a0, a1};
                v2f B = {b0, b1};
                acc = __builtin_amdgcn_wmma_f32_16x16x4_f32(
                    false, A, false, B, (short)0, acc, false, false);
            }
            __syncthreads();   // buffer bi free for reuse before next issue
        }
    }

    // D layout: vgpr j -> M = j (lanes 0-15) or j+8 (lanes 16-31), N = lane&15
#pragma unroll
    for (int j = 0; j < 8; ++j) {
        const int m   = j + 8 * half;
        const int row = base + m;
        if (row < n_rows) {
            const int   cnt   = s_off[m + 1] - s_off[m];
            const float scale = 1.0f / (float)(cnt > 0 ? cnt : 1);
            out[(long)row * D_FEAT + colbase + nn] = acc[j] * scale;
        }
    }
}

extern "C" void kernel_launch(void* const* d_in, const int* in_sizes, int n_in,
                              void* d_out, int out_size, void* d_ws, size_t ws_size,
                              hipStream_t stream) {
    const float* features  = (const float*)d_in[0];
    const int*   neigh_ids = (const int*)d_in[1];
    const int*   row_ids   = (const int*)d_in[2];
    float* out = (float*)d_out;

    const int n_edges = in_sizes[1];
    const int n_rows  = out_size / D_FEAT;
    const int n_nodes = in_sizes[0] / D_FEAT;
    const int n_tiles = (n_rows + 15) / 16;

    mean_agg_tdm_wmma<<<n_tiles, 256, 0, stream>>>(features, neigh_ids, row_ids, out,
                                                   n_edges, n_rows, n_nodes);
}